// ncc_test_60670708023332
// MI455X (gfx1250) — compile-verified
//
#include <hip/hip_runtime.h>
#include <hip/hip_bf16.h>
#include <math.h>

#define PSZ 256
#define NP  127
#define LL  16129   /* 127*127 */
#define BB  2
#define KK  4
#define NPL 12      /* K planes padded 9 -> 12 (zero planes 9..11) */
#define NCHUNK 1009 /* ceil(LL/16) */

typedef __attribute__((ext_vector_type(2))) float v2f;
typedef __attribute__((ext_vector_type(8))) float v8f;

// ---------------- K1: gray proxies ----------------
__global__ void k_gray(const float* __restrict__ x, const float* __restrict__ ref,
                       float* __restrict__ xg, float* __restrict__ rg) {
    int t = blockIdx.x * blockDim.x + threadIdx.x;
    if (t >= BB * PSZ * PSZ) return;
    int b  = t / (PSZ * PSZ);
    int ij = t % (PSZ * PSZ);
    int i = ij / PSZ, j = ij % PSZ;
    const float* xb = x + (size_t)b * 3 * PSZ * PSZ;
    float mx = -INFINITY;
    for (int di = -1; di <= 1; ++di) {
        int ii = i + di; if (ii < 0 || ii >= PSZ) continue;
        for (int dj = -1; dj <= 1; ++dj) {
            int jj = j + dj; if (jj < 0 || jj >= PSZ) continue;
            int o = ii * PSZ + jj;
            float cm = fmaxf(xb[o], fmaxf(xb[PSZ*PSZ + o], xb[2*PSZ*PSZ + o]));
            mx = fmaxf(mx, cm);
        }
    }
    float mn = fminf(xb[ij], fminf(xb[PSZ*PSZ + ij], xb[2*PSZ*PSZ + ij]));
    xg[t] = mx - mn;
    const float* rb = ref + (size_t)b * 3 * PSZ * PSZ;
    rg[t] = (rb[ij] + rb[PSZ*PSZ + ij] + rb[2*PSZ*PSZ + ij]) / 3.0f;
}

// ---------------- K2: unfold 3x3 stride 2 into padded [B][12][L] ----------------
__global__ void k_unfold(const float* __restrict__ xg, const float* __restrict__ rg,
                         float* __restrict__ ip, float* __restrict__ rp) {
    int t = blockIdx.x * blockDim.x + threadIdx.x;
    if (t >= BB * NPL * LL) return;
    int b = t / (NPL * LL);
    int r = t % (NPL * LL);
    int p = r / LL, l = r % LL;
    if (p >= 9) { ip[t] = 0.0f; rp[t] = 0.0f; return; }   // zero pad planes
    int row = 2 * (l / NP) + p / 3;
    int col = 2 * (l % NP) + p % 3;
    int g = (b * PSZ + row) * PSZ + col;
    ip[t] = xg[g];
    rp[t] = rg[g];
}

// ---------------- K3: per-(b,p) patch means (padded to 12, zeros for p>=9) ------
__global__ void k_mean(const float* __restrict__ ip, const float* __restrict__ rp,
                       float* __restrict__ mu_in, float* __restrict__ mu_ref) {
    __shared__ float s[256];
    int bp = blockIdx.x;                                   // 0..23 = b*12+p
    int p  = bp % NPL;
    const float* src = (blockIdx.y == 0) ? ip : rp;
    float*       dst = (blockIdx.y == 0) ? mu_in : mu_ref;
    if (p >= 9) { if (threadIdx.x == 0) dst[bp] = 0.0f; return; }
    const float* base = src + (size_t)bp * LL;
    float acc = 0.0f;
    for (int i = threadIdx.x; i < LL; i += 256) acc += base[i];
    s[threadIdx.x] = acc; __syncthreads();
    for (int st = 128; st > 0; st >>= 1) {
        if ((int)threadIdx.x < st) s[threadIdx.x] += s[threadIdx.x + st];
        __syncthreads();
    }
    if (threadIdx.x == 0) dst[bp] = s[0] / (float)LL;
}

// ---------------- K4: 1/||ref_patch[:,m]|| (uncentered) ----------------
__global__ void k_invn(const float* __restrict__ rp, float* __restrict__ invn) {
    int t = blockIdx.x * blockDim.x + threadIdx.x;
    if (t >= BB * LL) return;
    int b = t / LL, m = t % LL;
    const float* base = rp + (size_t)b * NPL * LL + m;
    float s = 0.0f;
#pragma unroll
    for (int p = 0; p < 9; ++p) { float v = base[(size_t)p * LL]; s += v * v; }
    invn[t] = 1.0f / sqrtf(s);
}

// ---- streaming insert: m strictly increasing per lane => strict value compare ----
__device__ __forceinline__ void top4_stream(float v, int m,
    float& v0, int& i0, float& v1, int& i1,
    float& v2, int& i2, float& v3, int& i3) {
    if (v > v3) {
        if (v > v0)      { v3=v2; i3=i2; v2=v1; i2=i1; v1=v0; i1=i0; v0=v; i0=m; }
        else if (v > v1) { v3=v2; i3=i2; v2=v1; i2=i1; v1=v;  i1=m; }
        else if (v > v2) { v3=v2; i3=i2; v2=v;  i2=m; }
        else             { v3=v;  i3=m; }
    }
}

// ---- merge insert: full jax tie-break (value desc, index asc) ----
__device__ __forceinline__ void top4_merge(float v, int m,
    float& v0, int& i0, float& v1, int& i1,
    float& v2, int& i2, float& v3, int& i3) {
    if (!((v > v3) || (v == v3 && m < i3))) return;
    if ((v > v0) || (v == v0 && m < i0)) {
        v3=v2; i3=i2; v2=v1; i2=i1; v1=v0; i1=i0; v0=v; i0=m;
    } else if ((v > v1) || (v == v1 && m < i1)) {
        v3=v2; i3=i2; v2=v1; i2=i1; v1=v; i1=m;
    } else if ((v > v2) || (v == v2 && m < i2)) {
        v3=v2; i3=i2; v2=v; i2=m;
    } else {
        v3=v; i3=m;
    }
}

// ---------------- K5: fused NCC GEMM (WMMA f32 16x16x4) + streaming top-4 --------
__global__ __launch_bounds__(32) void k_ncc_topk(
    const float* __restrict__ ip, const float* __restrict__ rp,
    const float* __restrict__ mu_in, const float* __restrict__ mu_ref,
    const float* __restrict__ invn, int* __restrict__ idxo) {
    __shared__ float tile[16][17];   // padded: conflict-free row reads
    __shared__ float mv[16][4];
    __shared__ int   msi[16][4];

    const int b     = blockIdx.y;
    const int lbase = blockIdx.x * 16;
    const int lane  = threadIdx.x;       // single wave32
    const int nlo   = lane & 15;
    const int half  = lane >> 4;
    const int kb    = half * 2;          // lanes 0-15: K 0,1 ; lanes 16-31: K 2,3

    const float* ipb = ip + (size_t)b * NPL * LL;
    const float* rpb = rp + (size_t)b * NPL * LL;
    const float* muI = mu_in + b * NPL;
    const float* muR = mu_ref + b * NPL;
    const float* inb = invn + (size_t)b * LL;

    const int row  = lbase + nlo;
    const bool rowv = (row < LL);

    // A fragments: 16 rows x K=12 (planes 9..11 are zero in memory, mu padded 0)
    v2f a0, a1, a2;
    a0.x = rowv ? (ipb[(size_t)(kb+0)*LL + row] - muI[kb+0]) : 0.0f;
    a0.y = rowv ? (ipb[(size_t)(kb+1)*LL + row] - muI[kb+1]) : 0.0f;
    a1.x = rowv ? (ipb[(size_t)(kb+4)*LL + row] - muI[kb+4]) : 0.0f;
    a1.y = rowv ? (ipb[(size_t)(kb+5)*LL + row] - muI[kb+5]) : 0.0f;
    a2.x = rowv ? (ipb[(size_t)(kb+8)*LL + row] - muI[kb+8]) : 0.0f;
    a2.y = rowv ? (ipb[(size_t)(kb+9)*LL + row] - muI[kb+9]) : 0.0f;

    float v0 = -INFINITY, v1 = -INFINITY, v2v = -INFINITY, v3 = -INFINITY;
    int   i0 = 0x7fffffff, i1 = 0x7fffffff, i2v = 0x7fffffff, i3 = 0x7fffffff;

    // ---- main loop: all 16 columns valid, no guards ----
    for (int ch = 0; ch < NCHUNK - 1; ++ch) {
        const int mb  = ch * 16;
        const int col = mb + nlo;
        v2f b0, b1, b2;
        b0.x = rpb[(size_t)(kb+0)*LL + col] - muR[kb+0];
        b0.y = rpb[(size_t)(kb+1)*LL + col] - muR[kb+1];
        b1.x = rpb[(size_t)(kb+4)*LL + col] - muR[kb+4];
        b1.y = rpb[(size_t)(kb+5)*LL + col] - muR[kb+5];
        b2.x = rpb[(size_t)(kb+8)*LL + col] - muR[kb+8];
        b2.y = rpb[(size_t)(kb+9)*LL + col] - muR[kb+9];
        const float sc = inb[col];        // per-column scale; N == lane&15 for all C regs

        v8f cc = {};
        cc = __builtin_amdgcn_wmma_f32_16x16x4_f32(false, a0, false, b0, (short)0, cc, false, false);
        cc = __builtin_amdgcn_wmma_f32_16x16x4_f32(false, a1, false, b1, (short)0, cc, false, false);
        cc = __builtin_amdgcn_wmma_f32_16x16x4_f32(false, a2, false, b2, (short)0, cc, false, false);

#pragma unroll
        for (int v = 0; v < 8; ++v) tile[v + half * 8][nlo] = cc[v] * sc;
        __syncthreads();

        // batched LDS reads, then max-gate
        float val[8];
#pragma unroll
        for (int j = 0; j < 8; ++j) val[j] = tile[nlo][half * 8 + j];
        float mx = fmaxf(fmaxf(fmaxf(val[0], val[1]), fmaxf(val[2], val[3])),
                         fmaxf(fmaxf(val[4], val[5]), fmaxf(val[6], val[7])));
        if (mx > v3) {
#pragma unroll
            for (int j = 0; j < 8; ++j)
                top4_stream(val[j], mb + half * 8 + j, v0, i0, v1, i1, v2v, i2v, v3, i3);
        }
        __syncthreads();
    }

    // ---- peeled final chunk (guards only here) ----
    {
        const int mb  = (NCHUNK - 1) * 16;
        const int col = mb + nlo;
        const bool cv = (col < LL);
        v2f b0, b1, b2;
        b0.x = cv ? (rpb[(size_t)(kb+0)*LL + col] - muR[kb+0]) : 0.0f;
        b0.y = cv ? (rpb[(size_t)(kb+1)*LL + col] - muR[kb+1]) : 0.0f;
        b1.x = cv ? (rpb[(size_t)(kb+4)*LL + col] - muR[kb+4]) : 0.0f;
        b1.y = cv ? (rpb[(size_t)(kb+5)*LL + col] - muR[kb+5]) : 0.0f;
        b2.x = cv ? (rpb[(size_t)(kb+8)*LL + col] - muR[kb+8]) : 0.0f;
        b2.y = cv ? (rpb[(size_t)(kb+9)*LL + col] - muR[kb+9]) : 0.0f;
        const float sc = cv ? inb[col] : 0.0f;

        v8f cc = {};
        cc = __builtin_amdgcn_wmma_f32_16x16x4_f32(false, a0, false, b0, (short)0, cc, false, false);
        cc = __builtin_amdgcn_wmma_f32_16x16x4_f32(false, a1, false, b1, (short)0, cc, false, false);
        cc = __builtin_amdgcn_wmma_f32_16x16x4_f32(false, a2, false, b2, (short)0, cc, false, false);

#pragma unroll
        for (int v = 0; v < 8; ++v) tile[v + half * 8][nlo] = cc[v] * sc;
        __syncthreads();
#pragma unroll
        for (int j = 0; j < 8; ++j) {
            const int m = mb + half * 8 + j;
            if (m < LL) {
                float val = tile[nlo][half * 8 + j];
                top4_stream(val, m, v0, i0, v1, i1, v2v, i2v, v3, i3);
            }
        }
        __syncthreads();
    }

    // ---- merge the two half-wave lists (needs full tie-break comparator) ----
    if (half == 1) {
        mv[nlo][0] = v0;  msi[nlo][0] = i0;
        mv[nlo][1] = v1;  msi[nlo][1] = i1;
        mv[nlo][2] = v2v; msi[nlo][2] = i2v;
        mv[nlo][3] = v3;  msi[nlo][3] = i3;
    }
    __syncthreads();
    if (half == 0) {
#pragma unroll
        for (int t = 0; t < 4; ++t)
            top4_merge(mv[nlo][t], msi[nlo][t], v0, i0, v1, i1, v2v, i2v, v3, i3);
        if (rowv) {
            int* o = idxo + ((size_t)b * LL + row) * KK;
            o[0] = i0; o[1] = i1; o[2] = i2v; o[3] = i3;
        }
    }
}

// ---------------- K6: fold (overlap-add gather) + concat with x ----------------
__global__ void k_out(const float* __restrict__ x, const float* __restrict__ rp,
                      const int* __restrict__ idxo, float* __restrict__ out) {
    int t = blockIdx.x * blockDim.x + threadIdx.x;
    if (t >= BB * 7 * PSZ * PSZ) return;
    int b  = t / (7 * PSZ * PSZ);
    int r  = t % (7 * PSZ * PSZ);
    int ch = r / (PSZ * PSZ);
    int ij = r % (PSZ * PSZ);
    if (ch < 3) { out[t] = x[((size_t)b * 3 + ch) * PSZ * PSZ + ij]; return; }
    int ki = ch - 3;
    int i = ij / PSZ, j = ij % PSZ;
    float acc = 0.0f;
    for (int di = 0; di < 3; ++di) {
        int ri = i - di;
        if (ri < 0 || (ri & 1)) continue;
        int rr = ri >> 1; if (rr >= NP) continue;
        for (int dj = 0; dj < 3; ++dj) {
            int cj = j - dj;
            if (cj < 0 || (cj & 1)) continue;
            int cc = cj >> 1; if (cc >= NP) continue;
            int l = rr * NP + cc;
            int m = idxo[((size_t)b * LL + l) * KK + ki];
            acc += rp[((size_t)b * NPL + (3 * di + dj)) * LL + m];
        }
    }
    out[t] = acc;
}

// ---------------- launch ----------------
extern "C" void kernel_launch(void* const* d_in, const int* in_sizes, int n_in,
                              void* d_out, int out_size, void* d_ws, size_t ws_size,
                              hipStream_t stream) {
    const float* x   = (const float*)d_in[0];
    const float* ref = (const float*)d_in[1];
    float* out = (float*)d_out;

    float* ws = (float*)d_ws;
    float* xg     = ws;                            // 2*65536
    float* rg     = xg + BB * PSZ * PSZ;           // 2*65536
    float* ip     = rg + BB * PSZ * PSZ;           // 2*12*LL
    float* rp     = ip + (size_t)BB * NPL * LL;    // 2*12*LL
    float* mu_in  = rp + (size_t)BB * NPL * LL;    // 24
    float* mu_ref = mu_in + BB * NPL;              // 24
    float* invn   = mu_ref + BB * NPL;             // 2*LL
    int*   idxo   = (int*)(invn + (size_t)BB * LL);// 2*LL*4 ints

    {   // K1
        int n = BB * PSZ * PSZ;
        k_gray<<<(n + 255) / 256, 256, 0, stream>>>(x, ref, xg, rg);
    }
    {   // K2 (also zero-fills pad planes every call)
        int n = BB * NPL * LL;
        k_unfold<<<(n + 255) / 256, 256, 0, stream>>>(xg, rg, ip, rp);
    }
    {   // K3
        k_mean<<<dim3(BB * NPL, 2), 256, 0, stream>>>(ip, rp, mu_in, mu_ref);
    }
    {   // K4
        int n = BB * LL;
        k_invn<<<(n + 255) / 256, 256, 0, stream>>>(rp, invn);
    }
    {   // K5: WMMA GEMM + fused top-4
        k_ncc_topk<<<dim3(NCHUNK, BB), 32, 0, stream>>>(ip, rp, mu_in, mu_ref, invn, idxo);
    }
    {   // K6
        int n = BB * 7 * PSZ * PSZ;
        k_out<<<(n + 255) / 256, 256, 0, stream>>>(x, rp, idxo, out);
    }
}